// MipNerfModel_61031485276506
// MI455X (gfx1250) — compile-verified
//
#include <hip/hip_runtime.h>

#define NS 128                 // NUM_SAMPLES
#define NT 129                 // NS + 1
#define F32_EPS_C 1.1920928955078125e-07f
#define RESAMPLE_PAD 0.01f

#if defined(__has_builtin)
#if __has_builtin(__builtin_amdgcn_global_load_async_to_lds_b32)
#define USE_ASYNC_LDS 1
#endif
#endif

typedef __attribute__((address_space(1))) int as1_int;
typedef __attribute__((address_space(3))) int as3_int;

__device__ __forceinline__ void frustum_gaussian(
    float t0, float t1,
    float dx, float dy, float dz,
    float ox, float oy, float oz,
    float inv_dmag2, float r2,
    float* mean, float* cov)
{
    float mu  = 0.5f * (t0 + t1);
    float hw  = 0.5f * (t1 - t0);
    float mu2 = mu * mu;
    float hw2 = hw * hw;
    float den = 3.0f * mu2 + hw2;
    float inv_den = 1.0f / den;
    float t_mean = mu + 2.0f * mu * hw2 * inv_den;
    float t_var  = hw2 * (1.0f / 3.0f)
                 - (4.0f / 15.0f) * (hw2 * hw2 * (12.0f * mu2 - hw2) * inv_den * inv_den);
    float r_var  = r2 * (0.25f * mu2 + (5.0f / 12.0f) * hw2
                 - (4.0f / 15.0f) * hw2 * hw2 * inv_den);

    float d[3] = {dx, dy, dz};
    float o[3] = {ox, oy, oz};
#pragma unroll
    for (int k = 0; k < 3; ++k) {
        float dk2 = d[k] * d[k];
        mean[k] = fmaf(d[k], t_mean, o[k]);
        cov[k]  = t_var * dk2 + r_var * (1.0f - dk2 * inv_dmag2);
    }
}

__global__ void __launch_bounds__(128) mipnerf_resample_kernel(
    const float* __restrict__ origins,     // [B,3]
    const float* __restrict__ directions,  // [B,3]
    const float* __restrict__ radii,       // [B,1]
    const float* __restrict__ nearv,       // [B,1]
    const float* __restrict__ farv,        // [B,1]
    const float* __restrict__ weights,     // [B,NS]
    float* __restrict__ out,
    int B)
{
    const int ray = blockIdx.x;
    const int tid = threadIdx.x;
    if (ray >= B) return;

    __shared__ float s_w[NS];
    __shared__ float s_tc[NT];
    __shared__ float s_scan[NS];
    __shared__ float s_cdf[NT];
    __shared__ float s_tf[NT];

    // output section offsets (tuple concat order)
    const size_t O_tc = 0;
    const size_t O_mc = (size_t)B * NT;
    const size_t O_cc = O_mc + (size_t)B * NS * 3;
    const size_t O_tf = O_cc + (size_t)B * NS * 3;
    const size_t O_mf = O_tf + (size_t)B * NT;
    const size_t O_cf = O_mf + (size_t)B * NS * 3;

    // ---- stage this ray's weights into LDS (async, overlapped with t_coarse) ----
    {
        const float* gp = weights + (size_t)ray * NS + tid;
#if USE_ASYNC_LDS
        __builtin_amdgcn_global_load_async_to_lds_b32(
            (as1_int*)(void*)gp, (as3_int*)&s_w[tid], 0, 0);
#else
        s_w[tid] = *gp;
#endif
    }

    // ---- per-ray uniforms (scalar-cached) ----
    const float nr  = nearv[ray];
    const float fr  = farv[ray];
    const float rad = radii[ray];
    const float dx = directions[(size_t)ray * 3 + 0];
    const float dy = directions[(size_t)ray * 3 + 1];
    const float dz = directions[(size_t)ray * 3 + 2];
    const float ox = origins[(size_t)ray * 3 + 0];
    const float oy = origins[(size_t)ray * 3 + 1];
    const float oz = origins[(size_t)ray * 3 + 2];
    const float dmag2 = fmaxf(dx * dx + dy * dy + dz * dz, 1e-10f);
    const float inv_dmag2 = 1.0f / dmag2;
    const float r2 = rad * rad;

    // ---- t_coarse: near*(1-t) + far*t, t = i/128 ----
    for (int i = tid; i < NT; i += NS) {
        float t  = (float)i * (1.0f / (float)NS);
        float tc = nr * (1.0f - t) + fr * t;
        s_tc[i] = tc;
        out[O_tc + (size_t)ray * NT + i] = tc;
    }

#if USE_ASYNC_LDS
#if defined(__has_builtin) && __has_builtin(__builtin_amdgcn_s_wait_asynccnt)
    __builtin_amdgcn_s_wait_asynccnt(0);
#else
    asm volatile("s_wait_asynccnt 0" ::: "memory");
#endif
#endif
    __syncthreads();

    // ---- weight blur: wblur[i] = 0.5*(max(w[i-1],w[i]) + max(w[i],w[i+1])) + pad ----
    {
        float wm1 = s_w[tid > 0 ? tid - 1 : 0];
        float w0  = s_w[tid];
        float wp1 = s_w[tid < NS - 1 ? tid + 1 : NS - 1];
        float wb  = 0.5f * (fmaxf(wm1, w0) + fmaxf(w0, wp1)) + RESAMPLE_PAD;
        s_scan[tid] = wb;
    }
    __syncthreads();

    // ---- inclusive prefix scan over wblur (Hillis-Steele, 7 steps) ----
#pragma unroll
    for (int off = 1; off < NS; off <<= 1) {
        float v = (tid >= off) ? s_scan[tid - off] : 0.0f;
        __syncthreads();
        s_scan[tid] += v;
        __syncthreads();
    }

    const float S       = s_scan[NS - 1];                 // total weight
    const float pad     = fmaxf(1e-5f - S, 0.0f);
    const float wsum    = S + pad;
    const float inv_ws  = 1.0f / wsum;
    const float pad_per = pad * (1.0f / (float)NS);

    // ---- cdf: [0, min(cumsum(pdf[:127]),1), 1]  -> 129 entries ----
    if (tid == 0) { s_cdf[0] = 0.0f; s_cdf[NS] = 1.0f; }
    if (tid < NS - 1) {
        float c = (s_scan[tid] + (float)(tid + 1) * pad_per) * inv_ws;
        s_cdf[tid + 1] = fminf(c, 1.0f);
    }
    __syncthreads();

    // ---- inverse-CDF sampling: searchsorted of u into cdf, lerp bins ----
    const float ustep = (1.0f - F32_EPS_C) * (1.0f / (float)NS);
    for (int s = tid; s < NT; s += NS) {
        float u = (float)s * ustep;
        // largest k in [0,128) with cdf[k] <= u  (cdf[0]=0<=u, cdf[128]=1>u)
        int lo = 0, hi = NS;
        while (hi - lo > 1) {
            int mid = (lo + hi) >> 1;
            if (s_cdf[mid] <= u) lo = mid; else hi = mid;
        }
        float g0 = s_cdf[lo], g1 = s_cdf[lo + 1];
        float b0 = s_tc[lo],  b1 = s_tc[lo + 1];
        float tt = (u - g0) / (g1 - g0);
        if (tt != tt) tt = 0.0f;              // nan_to_num (NaN -> 0)
        tt = fminf(fmaxf(tt, 0.0f), 1.0f);    // clip [0,1] (also clamps +/-inf)
        float tf = fmaf(tt, b1 - b0, b0);
        s_tf[s] = tf;
        out[O_tf + (size_t)ray * NT + s] = tf;
    }
    __syncthreads();

    // ---- Gaussians for coarse and fine intervals (thread i -> interval i) ----
    {
        float mean[3], cov[3];
        const size_t base = (size_t)ray * NS * 3 + (size_t)tid * 3;

        frustum_gaussian(s_tc[tid], s_tc[tid + 1], dx, dy, dz, ox, oy, oz,
                         inv_dmag2, r2, mean, cov);
#pragma unroll
        for (int k = 0; k < 3; ++k) {
            out[O_mc + base + k] = mean[k];
            out[O_cc + base + k] = cov[k];
        }

        frustum_gaussian(s_tf[tid], s_tf[tid + 1], dx, dy, dz, ox, oy, oz,
                         inv_dmag2, r2, mean, cov);
#pragma unroll
        for (int k = 0; k < 3; ++k) {
            out[O_mf + base + k] = mean[k];
            out[O_cf + base + k] = cov[k];
        }
    }
}

extern "C" void kernel_launch(void* const* d_in, const int* in_sizes, int n_in,
                              void* d_out, int out_size, void* d_ws, size_t ws_size,
                              hipStream_t stream) {
    (void)n_in; (void)out_size; (void)d_ws; (void)ws_size;
    const float* origins    = (const float*)d_in[0];
    const float* directions = (const float*)d_in[1];
    const float* radii      = (const float*)d_in[2];
    const float* nearv      = (const float*)d_in[3];
    const float* farv       = (const float*)d_in[4];
    const float* weights    = (const float*)d_in[5];
    float* out = (float*)d_out;

    const int B = in_sizes[2];  // radii is [B,1]
    mipnerf_resample_kernel<<<dim3(B), dim3(NS), 0, stream>>>(
        origins, directions, radii, nearv, farv, weights, out, B);
}